// OneHotEmbedding_46136538693902
// MI455X (gfx1250) — compile-verified
//
#include <hip/hip_runtime.h>
#include <hip/hip_bf16.h>
#include <stdint.h>

// out[tok, e] = W[e, ids[tok]]   (pure column gather from W [EMBED, VOCAB])
// Memory-bound: ~154MB W read (random lines) + 25MB write => ~8us @ 23.3TB/s.
// W fits in the 192MB L2, so default RT temporal hints keep replays in-L2.

#define VOCAB 50257
#define EMBED 768

__global__ __launch_bounds__(256) void onehot_embed_gather_kernel(
    const int* __restrict__ ids,
    const float* __restrict__ W,
    float* __restrict__ out)
{
    __shared__ float smem[EMBED];   // 3 KB staging row per block

    const int      t   = threadIdx.x;
    const unsigned tok = blockIdx.x;
    // Block-uniform token id -> scalar load + scalar addressing.
    const unsigned id  = (unsigned)ids[tok];

    // 32-bit LDS byte address of smem (wave-relative), for async-LDS ops.
    const unsigned lds_base =
        (unsigned)(unsigned long long)(__attribute__((address_space(3))) float*)smem;

    // ---- Phase 1: async-gather the strided column W[:, id] into LDS ----
    // 768 elements, 3 per thread. Each lane issues its own 4B async copy:
    // memory -> LDS directly, no VGPR round trip. Tracked by ASYNCcnt.
#pragma unroll
    for (int k = 0; k < 3; ++k) {
        const unsigned e    = (unsigned)t + 256u * k;
        const float*   gp   = W + ((unsigned long long)e * VOCAB + id);
        const unsigned loff = lds_base + e * 4u;
        asm volatile("global_load_async_to_lds_b32 %0, %1, off"
                     :
                     : "v"(loff), "v"(gp)
                     : "memory");
    }
    // Wait for this wave's async copies, then sync the workgroup so every
    // wave sees the fully-populated LDS row.
    asm volatile("s_wait_asynccnt 0x0" ::: "memory");
    __syncthreads();

    // ---- Phase 2: stream the contiguous 3KB row out with async b128 stores ----
    // 768 floats = 192 x 16B. LDS -> global directly.
    if (t < 192) {
        float*         op   = out + ((unsigned long long)tok * EMBED + 4u * (unsigned)t);
        const unsigned loff = lds_base + 16u * (unsigned)t;
        asm volatile("global_store_async_from_lds_b128 %0, %1, off"
                     :
                     : "v"(op), "v"(loff)
                     : "memory");
    }
    // S_ENDPGM implicitly waits for all counters, but be explicit: LDS must
    // stay live until the async stores have drained.
    asm volatile("s_wait_asynccnt 0x0" ::: "memory");
}

extern "C" void kernel_launch(void* const* d_in, const int* in_sizes, int n_in,
                              void* d_out, int out_size, void* d_ws, size_t ws_size,
                              hipStream_t stream)
{
    const int*   ids = (const int*)d_in[0];    // [BATCH*SEQ] int32 token ids
    const float* W   = (const float*)d_in[1];  // [EMBED, VOCAB] f32
    float*       out = (float*)d_out;          // [BATCH*SEQ, EMBED] f32

    const int ntok = in_sizes[0];              // BATCH*SEQ = 8192
    onehot_embed_gather_kernel<<<ntok, 256, 0, stream>>>(ids, W, out);
}